// CRF_66597762892207
// MI455X (gfx1250) — compile-verified
//
#include <hip/hip_runtime.h>
#include <math.h>

#define B_DIM 64
#define S_DIM 512
#define H_DIM 1024
#define K_DIM 7
#define KP    16   // K padded to WMMA N=16

typedef __attribute__((ext_vector_type(2))) float v2f;
typedef __attribute__((ext_vector_type(8))) float v8f;

// ---------------------------------------------------------------------------
// Kernel 1: emissions[b,s,k] = x[b,s,:] @ W[:,k] + bias[k]  via f32 WMMA.
// One wave per 16-row tile of the (B*S, H) x matrix; 8 waves per block.
// A-frag: lane m(<16) holds x[row0+m][h + 0..1]; lanes 16..31 hold K=2..3.
// B-frag: from LDS-staged W padded to 16 columns (cols 7..15 are zero).
// ---------------------------------------------------------------------------
__global__ __launch_bounds__(256) void emissions_wmma_kernel(
    const float* __restrict__ x, const float* __restrict__ W,
    const float* __restrict__ bias, float* __restrict__ em)
{
    __shared__ float Wp[H_DIM * KP];   // 64 KB of the WGP's 320 KB LDS
    const int tid = threadIdx.x;
    for (int i = tid; i < H_DIM * KP; i += 256) {
        int h = i >> 4, n = i & 15;
        Wp[i] = (n < K_DIM) ? W[h * K_DIM + n] : 0.0f;
    }
    __syncthreads();

    const int lane  = tid & 31;
    const int wave  = tid >> 5;
    const int m     = lane & 15;      // row within tile (A) / column (B, C/D)
    const int khalf = lane >> 4;      // 0: K=0..1,  1: K=2..3
    const int row0  = (blockIdx.x * 8 + wave) * 16;

    const float* xr   = x  + (size_t)(row0 + m) * H_DIM + khalf * 2;
    const float* wcol = Wp + khalf * 2 * KP + m;

    v8f acc = {};
#pragma unroll 4
    for (int h = 0; h < H_DIM; h += 4) {
        v2f a = *(const v2f*)(xr + h);            // global_load_b64, 8B aligned
        v2f bb;
        bb.x = wcol[h * KP];                      // B[K=khalf*2    ][n=m]
        bb.y = wcol[h * KP + KP];                 // B[K=khalf*2 + 1][n=m]
        acc = __builtin_amdgcn_wmma_f32_16x16x4_f32(
                  false, a, false, bb, (short)0, acc, false, false);
    }

    // D layout: vgpr i, lanes 0-15 -> M=i, lanes 16-31 -> M=8+i; N = lane&15
    if (m < K_DIM) {
        float bv = bias[m];
#pragma unroll
        for (int i = 0; i < 8; ++i) {
            int r = row0 + khalf * 8 + i;
            em[(size_t)r * K_DIM + m] = acc[i] + bv;
        }
    }
}

// ---------------------------------------------------------------------------
// Kernel 2: per-batch CRF forward recurrence + numerator.
// One wave32 per batch; lane k (< 7) owns state k of alpha.
// ---------------------------------------------------------------------------
__global__ __launch_bounds__(32) void crf_forward_kernel(
    const float* __restrict__ em, const long long* __restrict__ gt,
    const unsigned char* __restrict__ mask,
    const float* __restrict__ start_trans, const float* __restrict__ end_trans,
    const float* __restrict__ trans, float* __restrict__ llh_out)
{
    const int b    = blockIdx.x;
    const int lane = threadIdx.x;
    const float*        emb = em   + (size_t)b * S_DIM * K_DIM;
    const long long*    gtb = gt   + (size_t)b * S_DIM;
    const unsigned char* mb = mask + (size_t)b * S_DIM;

    const int k = (lane < K_DIM) ? lane : 0;
    float Tcol[K_DIM];
#pragma unroll
    for (int j = 0; j < K_DIM; ++j) Tcol[j] = trans[j * K_DIM + k];

    float alpha = start_trans[k] + emb[k];        // t = 0

    // numerator partial + mask count, strided over t across lanes
    float numpart = 0.0f;
    int   cnt     = 0;
    for (int t = lane; t < S_DIM; t += 32) {
        cnt += mb[t] ? 1 : 0;
        if (t >= 1) {
            float mf = mb[t] ? 1.0f : 0.0f;
            int gp = (int)gtb[t - 1], gc = (int)gtb[t];
            numpart += (trans[gp * K_DIM + gc] + emb[t * K_DIM + gc]) * mf;
        }
    }
    for (int off = 16; off > 0; off >>= 1) {
        numpart += __shfl_down(numpart, off);
        cnt     += __shfl_down(cnt, off);
    }

    // sequential alpha recurrence (latency chain, 511 steps)
    for (int t = 1; t < S_DIM; ++t) {
        float e = emb[t * K_DIM + k];
        float v[K_DIM];
        float mx = -INFINITY;
#pragma unroll
        for (int j = 0; j < K_DIM; ++j) {
            float s = __shfl(alpha, j) + Tcol[j];
            v[j] = s;
            mx = fmaxf(mx, s);
        }
        float ssum = 0.0f;
#pragma unroll
        for (int j = 0; j < K_DIM; ++j) ssum += __expf(v[j] - mx);
        float nxt = mx + __logf(ssum) + e;
        alpha = mb[t] ? nxt : alpha;
    }

    // denom = logsumexp_k(alpha_k + end_trans_k)
    float z  = (lane < K_DIM) ? alpha + end_trans[k] : -INFINITY;
    float zm = z;
    for (int off = 16; off > 0; off >>= 1) zm = fmaxf(zm, __shfl_down(zm, off));
    zm = __shfl(zm, 0);
    float ez = (lane < K_DIM) ? __expf(z - zm) : 0.0f;
    for (int off = 16; off > 0; off >>= 1) ez += __shfl_down(ez, off);

    if (lane == 0) {
        float denom = zm + __logf(ez);
        int g0 = (int)gtb[0];
        float num = start_trans[g0] + emb[g0] + numpart;
        int last = cnt - 1;
        num += end_trans[(int)gtb[last]];
        llh_out[b] = num - denom;
    }
}

// ---------------------------------------------------------------------------
// Kernel 3: out = -mean(llh) over 64 batches (single wave, deterministic).
// ---------------------------------------------------------------------------
__global__ __launch_bounds__(32) void reduce_mean_kernel(
    const float* __restrict__ llh, float* __restrict__ out)
{
    int lane = threadIdx.x;
    float v = llh[lane] + llh[lane + 32];
    for (int off = 16; off > 0; off >>= 1) v += __shfl_down(v, off);
    if (lane == 0) out[0] = -v / (float)B_DIM;
}

// ---------------------------------------------------------------------------
extern "C" void kernel_launch(void* const* d_in, const int* in_sizes, int n_in,
                              void* d_out, int out_size, void* d_ws, size_t ws_size,
                              hipStream_t stream) {
    (void)in_sizes; (void)n_in; (void)out_size; (void)ws_size;
    const float*         x    = (const float*)d_in[0];
    const long long*     gt   = (const long long*)d_in[1];
    const unsigned char* mask = (const unsigned char*)d_in[2];
    const float*         W    = (const float*)d_in[3];
    const float*         bias = (const float*)d_in[4];
    const float*         st   = (const float*)d_in[5];
    const float*         et   = (const float*)d_in[6];
    const float*         tr   = (const float*)d_in[7];

    float* em  = (float*)d_ws;                              // B*S*K f32 (~0.9 MB)
    float* llh = em + (size_t)B_DIM * S_DIM * K_DIM;        // 64 f32

    emissions_wmma_kernel<<<(B_DIM * S_DIM) / 128, 256, 0, stream>>>(x, W, bias, em);
    crf_forward_kernel<<<B_DIM, 32, 0, stream>>>(em, gt, mask, st, et, tr, llh);
    reduce_mean_kernel<<<1, 32, 0, stream>>>(llh, (float*)d_out);
}